// IRMEdgeCritic_39539468926979
// MI455X (gfx1250) — compile-verified
//
#include <hip/hip_runtime.h>
#include <math.h>

// ---------------------------------------------------------------------------
// IRM Edge Critic on gfx1250 (MI455X).
//
// All GEMMs in fp32 via V_WMMA_F32_16X16X4_F32 (matches fp32 reference; the
// workload is scatter/L2-atomic bound at ~45 GFLOP so low precision buys
// nothing). A tiles are staged memory->LDS with GLOBAL_LOAD_ASYNC_TO_LDS_B128
// (ASYNCcnt) — the CDNA5 async data-movement path. Each wave computes two
// 16x16 output tiles so every B fragment feeds two WMMAs.
// ---------------------------------------------------------------------------

typedef float v2f __attribute__((ext_vector_type(2)));
typedef float v8f __attribute__((ext_vector_type(8)));

__device__ __forceinline__ void atomicAddF(float* p, float v) {
    __hip_atomic_fetch_add(p, v, __ATOMIC_RELAXED, __HIP_MEMORY_SCOPE_AGENT);
}

// Async copy of 16 bytes global -> LDS (per active lane). Tracked by ASYNCcnt.
__device__ __forceinline__ void async_g2lds_b128(unsigned ldsOff, const float* g) {
    asm volatile("global_load_async_to_lds_b128 %0, %1, off"
                 :: "v"(ldsOff), "v"(g) : "memory");
}
__device__ __forceinline__ void async_wait0() {
    asm volatile("s_wait_asynccnt 0x0" ::: "memory");
}

// Zero a float region.
__global__ void zero_f32(float* __restrict__ p, long long n) {
    long long t = (long long)blockIdx.x * blockDim.x + threadIdx.x;
    if (t < n) p[t] = 0.0f;
}

// ---------------------------------------------------------------------------
// Node GEMM: out[32n, 64] = A[32n, K] @ W[K, 64]. One 32-row tile per block,
// 4 waves; each wave owns one 16-col strip and two 16-row accumulators so a
// B fragment is loaded once and used by two WMMAs.
// DUAL:   Y = acc (raw, for edge gathers), U = acc + bias (aggregation init)
// !DUAL:  Y = relu(acc + bias)
// RELU_IN: relu applied to A at fragment-read time (A staged raw via async).
// ---------------------------------------------------------------------------
template <int K, bool RELU_IN, bool DUAL>
__global__ __launch_bounds__(128) void gin_gemm(
    const float* __restrict__ A, const float* __restrict__ W,
    const float* __restrict__ bias, float* __restrict__ Y, float* __restrict__ U)
{
    constexpr int PITCH = K + 4;   // rows 16B-aligned; bank stride %64 == 4
    __shared__ float As[32 * PITCH];
    const int node0 = blockIdx.x * 32;
    const int tid = threadIdx.x;

    // Stage the 32xK A tile into LDS with async B128 copies.
    constexpr int C4 = K / 4;                       // 16B chunks per row
    for (int i = tid; i < 32 * C4; i += 128) {
        int r = i / C4, c = (i % C4) * 4;
        const float* g = A + (size_t)(node0 + r) * K + c;
        unsigned ldsOff = (unsigned)(uintptr_t)(As + r * PITCH + c);
        async_g2lds_b128(ldsOff, g);
    }
    async_wait0();
    __syncthreads();

    const int lane = tid & 31, wv = tid >> 5;
    const int lhalf = lane >> 4, lrow = lane & 15;
    const int col = wv * 16 + lrow;

    v8f acc0 = {0.f, 0.f, 0.f, 0.f, 0.f, 0.f, 0.f, 0.f};
    v8f acc1 = {0.f, 0.f, 0.f, 0.f, 0.f, 0.f, 0.f, 0.f};
    // A frag: lane<16 -> A[lrow][k..k+1]; lane>=16 -> A[lrow][k+2..k+3]
    const float* arow0 = As + lrow * PITCH + 2 * lhalf;
    const float* arow1 = arow0 + 16 * PITCH;
    // B frag: lane<16 -> W[k..k+1][col]; lane>=16 -> W[k+2..k+3][col]
    const float* wcol = W + col + (size_t)(2 * lhalf) * 64;
    for (int k = 0; k < K; k += 4) {
        v2f b, a0, a1;
        b.x = wcol[(size_t)k * 64];
        b.y = wcol[(size_t)(k + 1) * 64];
        a0.x = arow0[k];  a0.y = arow0[k + 1];
        a1.x = arow1[k];  a1.y = arow1[k + 1];
        if (RELU_IN) {
            a0.x = fmaxf(a0.x, 0.f); a0.y = fmaxf(a0.y, 0.f);
            a1.x = fmaxf(a1.x, 0.f); a1.y = fmaxf(a1.y, 0.f);
        }
        acc0 = __builtin_amdgcn_wmma_f32_16x16x4_f32(
            false, a0, false, b, (short)0, acc0, false, false);
        acc1 = __builtin_amdgcn_wmma_f32_16x16x4_f32(
            false, a1, false, b, (short)0, acc1, false, false);
    }

    const float bv = bias[col];
#pragma unroll
    for (int v = 0; v < 8; ++v) {
        int row = v + 8 * lhalf;                 // D layout: VGPR v -> M=v / M=v+8
        size_t i0 = (size_t)(node0 + row) * 64 + col;
        size_t i1 = (size_t)(node0 + 16 + row) * 64 + col;
        float v0 = acc0[v], v1 = acc1[v];
        if constexpr (DUAL) {
            Y[i0] = v0;  U[i0] = v0 + bv;
            Y[i1] = v1;  U[i1] = v1 + bv;
        } else {
            Y[i0] = fmaxf(v0 + bv, 0.0f);
            Y[i1] = fmaxf(v1 + bv, 0.0f);
        }
    }
}

// ---------------------------------------------------------------------------
// Edge aggregation: U[dst] += Y[src], 64-wide, 8 threads per edge.
// ---------------------------------------------------------------------------
__global__ __launch_bounds__(256) void scatter_agg(
    const float* __restrict__ Y, float* __restrict__ U,
    const long long* __restrict__ ei, int E)
{
    long long t = (long long)blockIdx.x * blockDim.x + threadIdx.x;
    if (t >= (long long)E * 8) return;
    int e = (int)(t >> 3);
    int f0 = ((int)t & 7) * 8;
    int s = (int)ei[e];
    int d = (int)ei[(size_t)E + e];
    const float* ys = Y + (size_t)s * 64 + f0;
    float* ud = U + (size_t)d * 64 + f0;
#pragma unroll
    for (int i = 0; i < 8; ++i) atomicAddF(ud + i, ys[i]);
}

// ---------------------------------------------------------------------------
// Edge scoring: 32 edges/block. pair[32,192] @ e_w1[192,64] via WMMA — 4 waves,
// each with one 16-col strip and two 16-edge accumulators (B reused 2x).
// Then relu -> LDS, 64-dot second layer, sigmoid, nw scatter.
// ---------------------------------------------------------------------------
__global__ __launch_bounds__(128) void edge_score(
    const float* __restrict__ h, const long long* __restrict__ ei,
    const float* __restrict__ e_w1, const float* __restrict__ e_b1,
    const float* __restrict__ e_w2, const float* __restrict__ e_b2,
    float* __restrict__ scores, float* __restrict__ nw, int E)
{
    constexpr int PP = 196;                      // 196%64==4 -> conflict free
    __shared__ float pairs[32 * PP];
    __shared__ float q[32 * 65];
    __shared__ int sid[32], did[32];
    const int e0 = blockIdx.x * 32;
    const int tid = threadIdx.x;

    if (tid < 32) {
        sid[tid] = (int)ei[e0 + tid];
        did[tid] = (int)ei[(size_t)E + e0 + tid];
    }
    __syncthreads();
    for (int i = tid; i < 32 * 8; i += 128) {
        int e = i >> 3, f0 = (i & 7) * 8;
        const float* hs = h + (size_t)sid[e] * 64 + f0;
        const float* hd = h + (size_t)did[e] * 64 + f0;
#pragma unroll
        for (int j = 0; j < 8; ++j) {
            float a = hs[j], b = hd[j];
            pairs[e * PP + f0 + j] = a;
            pairs[e * PP + 64 + f0 + j] = b;
            pairs[e * PP + 128 + f0 + j] = fabsf(a - b);
        }
    }
    __syncthreads();

    const int lane = tid & 31, wv = tid >> 5;
    const int lhalf = lane >> 4, lrow = lane & 15;
    const int col = wv * 16 + lrow;

    v8f acc0 = {0.f, 0.f, 0.f, 0.f, 0.f, 0.f, 0.f, 0.f};
    v8f acc1 = {0.f, 0.f, 0.f, 0.f, 0.f, 0.f, 0.f, 0.f};
    const float* arow0 = pairs + lrow * PP + 2 * lhalf;
    const float* arow1 = arow0 + 16 * PP;
    const float* wcol = e_w1 + col + (size_t)(2 * lhalf) * 64;
    for (int k = 0; k < 192; k += 4) {
        v2f b, a0, a1;
        b.x = wcol[(size_t)k * 64];
        b.y = wcol[(size_t)(k + 1) * 64];
        a0.x = arow0[k];  a0.y = arow0[k + 1];
        a1.x = arow1[k];  a1.y = arow1[k + 1];
        acc0 = __builtin_amdgcn_wmma_f32_16x16x4_f32(
            false, a0, false, b, (short)0, acc0, false, false);
        acc1 = __builtin_amdgcn_wmma_f32_16x16x4_f32(
            false, a1, false, b, (short)0, acc1, false, false);
    }
    const float bv = e_b1[col];
#pragma unroll
    for (int v = 0; v < 8; ++v) {
        int row = v + 8 * lhalf;
        q[row * 65 + col]        = fmaxf(acc0[v] + bv, 0.0f);
        q[(row + 16) * 65 + col] = fmaxf(acc1[v] + bv, 0.0f);
    }
    __syncthreads();

    if (tid < 32) {
        float s = e_b2[0];
        for (int f = 0; f < 64; ++f) s += q[tid * 65 + f] * e_w2[f];
        s = 1.0f / (1.0f + expf(-s));
        scores[e0 + tid] = s;
        atomicAddF(nw + sid[tid], s);
        atomicAddF(nw + did[tid], s);
    }
}

// ---------------------------------------------------------------------------
// Weighted pooling per graph: 16 threads/node (4 feats each).
// ---------------------------------------------------------------------------
__global__ __launch_bounds__(256) void pool_nodes(
    const float* __restrict__ h, const float* __restrict__ nw,
    const long long* __restrict__ batch,
    float* __restrict__ pooled, float* __restrict__ norm, int N)
{
    long long t = (long long)blockIdx.x * blockDim.x + threadIdx.x;
    if (t >= (long long)N * 16) return;
    int i = (int)(t >> 4);
    int f0 = ((int)t & 15) * 4;
    float w = nw[i] + 1e-6f;
    int g = (int)batch[i];
    const float* hp = h + (size_t)i * 64 + f0;
#pragma unroll
    for (int k = 0; k < 4; ++k) atomicAddF(pooled + g * 64 + f0 + k, hp[k] * w);
    if (f0 == 0) atomicAddF(norm + g, w);
}

__global__ void finalize_emb(const float* __restrict__ pooled,
                             const float* __restrict__ norm,
                             float* __restrict__ emb)
{
    int t = blockIdx.x * blockDim.x + threadIdx.x;   // G*64 = 4096
    if (t >= 64 * 64) return;
    int g = t >> 6;
    emb[t] = pooled[t] / fmaxf(norm[g], 1e-6f);
}

__global__ void domain_head(const float* __restrict__ emb,
                            const float* __restrict__ d_w,
                            const float* __restrict__ d_b,
                            float* __restrict__ logits)
{
    int t = blockIdx.x * blockDim.x + threadIdx.x;   // G*3 = 192
    if (t >= 64 * 3) return;
    int g = t / 3, d = t % 3;
    float s = d_b[d];
    for (int f = 0; f < 64; ++f) s += emb[(size_t)g * 64 + f] * d_w[f * 3 + d];
    logits[t] = s;
}

// ---------------------------------------------------------------------------
extern "C" void kernel_launch(void* const* d_in, const int* in_sizes, int n_in,
                              void* d_out, int out_size, void* d_ws, size_t ws_size,
                              hipStream_t stream)
{
    (void)n_in; (void)out_size; (void)ws_size;
    constexpr int IN_DIM = 256, H = 64, G = 64, D = 3;
    const int N = in_sizes[0] / IN_DIM;     // 100000 (multiple of 32)
    const int E = in_sizes[1] / 2;          // 1600000 (multiple of 32)

    const float*     x     = (const float*)d_in[0];
    const long long* ei    = (const long long*)d_in[1];
    const long long* batch = (const long long*)d_in[2];
    const float* g1_w1 = (const float*)d_in[3];
    const float* g1_b1 = (const float*)d_in[4];
    const float* g1_w2 = (const float*)d_in[5];
    const float* g1_b2 = (const float*)d_in[6];
    const float* g2_w1 = (const float*)d_in[7];
    const float* g2_b1 = (const float*)d_in[8];
    const float* g2_w2 = (const float*)d_in[9];
    const float* g2_b2 = (const float*)d_in[10];
    const float* e_w1  = (const float*)d_in[11];
    const float* e_b1  = (const float*)d_in[12];
    const float* e_w2  = (const float*)d_in[13];
    const float* e_b2  = (const float*)d_in[14];
    const float* d_w   = (const float*)d_in[15];
    const float* d_b   = (const float*)d_in[16];

    // Outputs (flat, in return order): scores[E], logits[G*D], h[N*H], emb[G*H]
    float* out_scores = (float*)d_out;
    float* out_logits = out_scores + E;
    float* out_h      = out_logits + (size_t)G * D;
    float* out_emb    = out_h + (size_t)N * H;

    // Workspace: Y[N*64], U[N*64], nw[N], pooled[G*64], norm[G]
    float* bufY   = (float*)d_ws;
    float* bufU   = bufY + (size_t)N * H;
    float* nw     = bufU + (size_t)N * H;
    float* pooled = nw + N;
    float* norm   = pooled + (size_t)G * H;

    const int nodeTiles = N / 32;   // 3125
    const int edgeTiles = E / 32;   // 50000

    // 0) zero nw / pooled / norm (d_ws is not re-poisoned between replays)
    {
        long long nz = (long long)N + (long long)G * H + G;
        zero_f32<<<(int)((nz + 255) / 256), 256, 0, stream>>>(nw, nz);
    }

    // 1) GIN layer 1
    gin_gemm<IN_DIM, false, true><<<nodeTiles, 128, 0, stream>>>(x, g1_w1, g1_b1, bufY, bufU);
    scatter_agg<<<(int)(((long long)E * 8 + 255) / 256), 256, 0, stream>>>(bufY, bufU, ei, E);
    gin_gemm<H, true, false><<<nodeTiles, 128, 0, stream>>>(bufU, g1_w2, g1_b2, out_h, nullptr);

    // 2) GIN layer 2 (h1 lives temporarily in out_h, overwritten by final h)
    gin_gemm<H, false, true><<<nodeTiles, 128, 0, stream>>>(out_h, g2_w1, g2_b1, bufY, bufU);
    scatter_agg<<<(int)(((long long)E * 8 + 255) / 256), 256, 0, stream>>>(bufY, bufU, ei, E);
    gin_gemm<H, true, false><<<nodeTiles, 128, 0, stream>>>(bufU, g2_w2, g2_b2, out_h, nullptr);

    // 3) edge scoring + nw scatter
    edge_score<<<edgeTiles, 128, 0, stream>>>(out_h, ei, e_w1, e_b1, e_w2, e_b2,
                                              out_scores, nw, E);

    // 4) weighted pooling, graph embedding, domain head
    pool_nodes<<<(int)(((long long)N * 16 + 255) / 256), 256, 0, stream>>>(
        out_h, nw, batch, pooled, norm, N);
    finalize_emb<<<(G * H + 255) / 256, 256, 0, stream>>>(pooled, norm, out_emb);
    domain_head<<<(G * D + 63) / 64, 64, 0, stream>>>(out_emb, d_w, d_b, out_logits);
}